// VectorQuantizer_79628693668055
// MI455X (gfx1250) — compile-verified
//
#include <hip/hip_runtime.h>
#include <hip/hip_bf16.h>
#include <math.h>

// ---------------------------------------------------------------------------
// VQ-VAE vector quantizer for MI455X (gfx1250, wave32, WMMA).
//   inputs:   d_in[0] = inputs   (16,64,64,64) f32 -> flat [N=65536, D=64]
//             d_in[1] = codebook (64,1024)     f32    [D=64, K=1024]
//   output d_out (f32), concatenated:
//             [0 .. 4194304)        ste (== quantized forward value)
//             [4194304]             perplexity
//             [4194305]             codebook_loss
//             [4194306]             commitment_loss (0.25 * same MSE)
//             [4194307 .. +65536)   encoding indices (as float)
//   workspace (floats): cnorm[1024] | counts[1024] | sse[1]   (~8.2 KB)
// ---------------------------------------------------------------------------

#define VQ_D      64
#define VQ_K      1024
#define VQ_N      65536          // 16*64*64*64 / 64
#define VQ_ELEMS  4194304        // N * D
#define MROWS     32             // rows per block in the distance kernel
#define ASTR      68             // padded LDS row stride (floats) -> conflict-free

#define OUT_PERP  4194304
#define OUT_CBL   4194305
#define OUT_CML   4194306
#define OUT_IDX   4194307

typedef float v2f __attribute__((ext_vector_type(2)));
typedef float v8f __attribute__((ext_vector_type(8)));

__device__ __forceinline__ unsigned f2ord(float f) {
    // monotonic float -> uint mapping (unsigned order == float order)
    unsigned u = __float_as_uint(f);
    return (u & 0x80000000u) ? ~u : (u ^ 0x80000000u);
}

// --------------------------------------------------------------------------
// Kernel 0: codebook column norms + zero accumulators.
// grid: 4 x 256  (1024 threads == K)
// --------------------------------------------------------------------------
__global__ __launch_bounds__(256) void vq_init(const float* __restrict__ cb,
                                               float* __restrict__ cnorm,
                                               float* __restrict__ counts,
                                               float* __restrict__ sse) {
    const int k = blockIdx.x * 256 + threadIdx.x;   // 0..1023
    float s = 0.0f;
#pragma unroll 8
    for (int d = 0; d < VQ_D; ++d) {
        float v = cb[(size_t)d * VQ_K + k];
        s += v * v;
    }
    cnorm[k]  = s;
    counts[k] = 0.0f;
    if (k == 0) sse[0] = 0.0f;
}

// --------------------------------------------------------------------------
// Kernel 1: distances + argmin via fp32 WMMA.
// grid: N/MROWS = 2048 blocks x 256 threads (8 waves).
// Each block: 32 rows x all 1024 codes. Wave w sweeps column tiles
// nt = w, w+8, ... (8 tiles of 16 codes); B fragment reused by 2 row tiles.
// --------------------------------------------------------------------------
__global__ __launch_bounds__(256) void vq_dist_argmin(const float* __restrict__ x,
                                                      const float* __restrict__ cb,
                                                      const float* __restrict__ cnorm,
                                                      float* __restrict__ idx_out,
                                                      float* __restrict__ counts) {
    __shared__ __align__(16) float Alds[MROWS * ASTR];
    __shared__ unsigned long long best64[MROWS];

    const int tid  = threadIdx.x;
    const int lane = tid & 31;
    const int wave = tid >> 5;
    const size_t rowbase = (size_t)blockIdx.x * MROWS;

    // Stage A-tile (32 rows x 64 f32) into LDS, coalesced float4 loads.
    for (int c = tid; c < MROWS * 16; c += 256) {
        const int r  = c >> 4;
        const int c4 = (c & 15) << 2;
        const float4 v = *(const float4*)(x + (rowbase + r) * VQ_D + c4);
        *(float4*)&Alds[r * ASTR + c4] = v;
    }
    if (tid < MROWS) best64[tid] = ~0ULL;
    __syncthreads();

    const int m16  = lane & 15;
    const int hi   = lane >> 4;        // 0: lanes 0-15, 1: lanes 16-31
    const int koff = hi << 1;          // fp32 WMMA A/B K-sublane offset: 0 or 2

    float bscore[2][8];
    int   bn[2][8];
#pragma unroll
    for (int t = 0; t < 2; ++t)
#pragma unroll
        for (int r = 0; r < 8; ++r) { bscore[t][r] = INFINITY; bn[t][r] = 0; }

    for (int nt = wave; nt < VQ_K / 16; nt += 8) {
        const int n  = (nt << 4) + m16;        // code index owned by this lane
        const float cn = cnorm[n];

        v8f c0 = {};
        v8f c1 = {};
#pragma unroll
        for (int kb = 0; kb < 16; ++kb) {      // 16 x (K=4) = D=64
            const int k0 = (kb << 2) + koff;
            v2f b;
            b.x = cb[(size_t)(k0    ) * VQ_K + n];
            b.y = cb[(size_t)(k0 + 1) * VQ_K + n];
            v2f a0 = *(const v2f*)&Alds[(m16     ) * ASTR + k0];
            v2f a1 = *(const v2f*)&Alds[(m16 + 16) * ASTR + k0];
            c0 = __builtin_amdgcn_wmma_f32_16x16x4_f32(false, a0, false, b,
                                                       (short)0, c0, false, false);
            c1 = __builtin_amdgcn_wmma_f32_16x16x4_f32(false, a1, false, b,
                                                       (short)0, c1, false, false);
        }

        // score = ||e_n||^2 - 2 x.e  (||x||^2 is row-constant, drop for argmin)
#pragma unroll
        for (int r = 0; r < 8; ++r) {
            const float s0 = fmaf(-2.0f, c0[r], cn);
            const float s1 = fmaf(-2.0f, c1[r], cn);
            if (s0 < bscore[0][r]) { bscore[0][r] = s0; bn[0][r] = n; }
            if (s1 < bscore[1][r]) { bscore[1][r] = s1; bn[1][r] = n; }
        }
    }

    // Merge lane-local winners: (ordered-score << 32) | n, ds_min_u64.
#pragma unroll
    for (int t = 0; t < 2; ++t)
#pragma unroll
        for (int r = 0; r < 8; ++r) {
            const int m = t * 16 + r + (hi << 3);   // C-layout: hi lanes hold M=r+8
            const unsigned long long p =
                ((unsigned long long)f2ord(bscore[t][r]) << 32) |
                (unsigned long long)(unsigned)bn[t][r];
            atomicMin(&best64[m], p);
        }
    __syncthreads();

    if (tid < MROWS) {
        const unsigned code = (unsigned)(best64[tid] & 0xFFFFFFFFu);
        idx_out[rowbase + tid] = (float)code;
        atomicAdd(&counts[code], 1.0f);
    }
}

// --------------------------------------------------------------------------
// Kernel 2: gather quantized rows, emit ste (= quantized), accumulate SSE.
// grid: ELEMS/256 = 16384 blocks x 256 threads.
// --------------------------------------------------------------------------
__global__ __launch_bounds__(256) void vq_quant(const float* __restrict__ x,
                                                const float* __restrict__ cb,
                                                const float* __restrict__ idx_f,
                                                float* __restrict__ ste,
                                                float* __restrict__ sse) {
    __shared__ float red[256];
    const size_t e   = (size_t)blockIdx.x * 256 + threadIdx.x;
    const size_t row = e >> 6;
    const int    d   = (int)(e & 63);
    const int    code = (int)idx_f[row];

    const float q  = cb[(size_t)d * VQ_K + code];   // codebook is L2-hot (256 KB)
    const float in = x[e];
    ste[e] = q;                                     // forward value of the STE

    const float diff = q - in;
    red[threadIdx.x] = diff * diff;
    __syncthreads();
#pragma unroll
    for (int s = 128; s > 0; s >>= 1) {
        if (threadIdx.x < s) red[threadIdx.x] += red[threadIdx.x + s];
        __syncthreads();
    }
    if (threadIdx.x == 0) atomicAdd(sse, red[0]);
}

// --------------------------------------------------------------------------
// Kernel 3: perplexity + losses. 1 block x 1024 threads.
// --------------------------------------------------------------------------
__global__ __launch_bounds__(1024) void vq_final(const float* __restrict__ counts,
                                                 const float* __restrict__ sse,
                                                 float* __restrict__ out) {
    __shared__ float red[1024];
    const int t = threadIdx.x;
    const float p = counts[t] * (1.0f / (float)VQ_N);
    red[t] = -p * logf(p + 1e-10f);
    __syncthreads();
#pragma unroll
    for (int s = 512; s > 0; s >>= 1) {
        if (t < s) red[t] += red[t + s];
        __syncthreads();
    }
    if (t == 0) {
        const float loss = sse[0] * (1.0f / (float)VQ_ELEMS);
        out[OUT_PERP] = expf(red[0]);
        out[OUT_CBL]  = loss;
        out[OUT_CML]  = 0.25f * loss;
    }
}

// --------------------------------------------------------------------------
extern "C" void kernel_launch(void* const* d_in, const int* in_sizes, int n_in,
                              void* d_out, int out_size, void* d_ws, size_t ws_size,
                              hipStream_t stream) {
    (void)in_sizes; (void)n_in; (void)out_size; (void)ws_size;

    const float* x  = (const float*)d_in[0];   // inputs  [N, D] flattened
    const float* cb = (const float*)d_in[1];   // codebook [D, K]
    float* out = (float*)d_out;

    float* cnorm  = (float*)d_ws;              // [1024]
    float* counts = cnorm + VQ_K;              // [1024]
    float* sse    = counts + VQ_K;             // [1]

    float* idx_out = out + OUT_IDX;            // indices live in d_out (as float)

    vq_init<<<VQ_K / 256, 256, 0, stream>>>(cb, cnorm, counts, sse);
    vq_dist_argmin<<<VQ_N / MROWS, 256, 0, stream>>>(x, cb, cnorm, idx_out, counts);
    vq_quant<<<VQ_ELEMS / 256, 256, 0, stream>>>(x, cb, idx_out, out, sse);
    vq_final<<<1, 1024, 0, stream>>>(counts, sse, out);
}